// Model_79748952752241
// MI455X (gfx1250) — compile-verified
//
#include <hip/hip_runtime.h>
#include <hip/hip_bf16.h>
#include <math.h>

typedef __attribute__((ext_vector_type(2))) float v2f;
typedef __attribute__((ext_vector_type(8))) float v8f;
typedef __attribute__((ext_vector_type(4))) int   v4i;

// pointer-to-AS-qualified-int4 types for the async global->LDS builtin
typedef __attribute__((address_space(1))) v4i* gas_v4i_ptr;
typedef __attribute__((address_space(3))) v4i* las_v4i_ptr;

#define SDIM 128
#define NDIM 256
#define DDIM 64
#define HEADS 4
#define HD 16
#define ROWS (SDIM * NDIM)   // 32768

#if defined(__has_builtin)
# if __has_builtin(__builtin_amdgcn_global_load_async_to_lds_b128) && \
     __has_builtin(__builtin_amdgcn_s_wait_asynccnt)
#  define USE_ASYNC_LDS 1
# endif
#endif

__device__ __forceinline__ v8f wmma_f32(v2f a, v2f b, v8f c) {
    return __builtin_amdgcn_wmma_f32_16x16x4_f32(false, a, false, b, (short)0, c,
                                                 false, false);
}

// 16-byte global -> LDS copy; uses CDNA5 async DMA path when available.
__device__ __forceinline__ void async_copy16(const float* g, float* l) {
#ifdef USE_ASYNC_LDS
    __builtin_amdgcn_global_load_async_to_lds_b128(
        (gas_v4i_ptr)(void*)g,
        (las_v4i_ptr)(void*)l,
        /*offset=*/0, /*cpol=*/0);
#else
    *(float4*)l = *(const float4*)g;
#endif
}

__device__ __forceinline__ void async_copy_wait() {
#ifdef USE_ASYNC_LDS
    __builtin_amdgcn_s_wait_asynccnt(0);
#endif
}

// ---------------------------------------------------------------------------
// Kernel 1: LayerNorm + 4 projections (q,k,v,g).  grid=2048 blocks of 128 thr.
// Each block: 16 rows of x. Wave w in {0..3} computes x@W_w (16x64) via WMMA.
// ---------------------------------------------------------------------------
__global__ __launch_bounds__(128)
void ln_qkvg_kernel(const float* __restrict__ msa,
                    const float* __restrict__ lng, const float* __restrict__ lnb,
                    const float* __restrict__ Wq, const float* __restrict__ Wk,
                    const float* __restrict__ Wv, const float* __restrict__ Wg,
                    const float* __restrict__ bg,
                    float* __restrict__ q, float* __restrict__ k,
                    float* __restrict__ v, float* __restrict__ g)
{
    __shared__ float xt[16][68];        // padded stride (gcd(68,64)=4 -> 16 banks)
    __shared__ float part[16][8][2];
    __shared__ float stats[16][2];

    const int tid  = threadIdx.x;
    const int row0 = blockIdx.x * 16;

    {   // partial sums: 8 threads per row, 8 elems per thread
        int r = tid >> 3, s8 = tid & 7;
        const float* src = msa + (size_t)(row0 + r) * DDIM + s8 * 8;
        float sum = 0.f, sq = 0.f;
        #pragma unroll
        for (int i = 0; i < 8; ++i) {
            float x = src[i];
            xt[r][s8 * 8 + i] = x;
            sum += x; sq += x * x;
        }
        part[r][s8][0] = sum; part[r][s8][1] = sq;
    }
    __syncthreads();
    if (tid < 16) {
        float sum = 0.f, sq = 0.f;
        #pragma unroll
        for (int i = 0; i < 8; ++i) { sum += part[tid][i][0]; sq += part[tid][i][1]; }
        float mu  = sum * (1.f / 64.f);
        float var = sq * (1.f / 64.f) - mu * mu;
        stats[tid][0] = mu;
        stats[tid][1] = rsqrtf(var + 1e-5f);
    }
    __syncthreads();
    {   // normalize in place
        int r = tid >> 3, s8 = tid & 7;
        float mu = stats[r][0], rs = stats[r][1];
        #pragma unroll
        for (int i = 0; i < 8; ++i) {
            int c = s8 * 8 + i;
            xt[r][c] = (xt[r][c] - mu) * rs * lng[c] + lnb[c];
        }
    }
    __syncthreads();

    const int wave = tid >> 5, lane = tid & 31;
    const int lr = lane & 15, hi = lane >> 4;
    const float* W   = (wave == 0) ? Wq : (wave == 1) ? Wk : (wave == 2) ? Wv : Wg;
    float*       out = (wave == 0) ? q  : (wave == 1) ? k  : (wave == 2) ? v  : g;

    #pragma unroll
    for (int nt = 0; nt < 4; ++nt) {
        v8f acc = {};
        #pragma unroll
        for (int kt = 0; kt < 16; ++kt) {
            int k0 = kt * 4 + hi * 2;
            v2f a; a.x = xt[lr][k0]; a.y = xt[lr][k0 + 1];
            v2f b; b.x = W[(size_t)k0 * DDIM + nt * 16 + lr];
                   b.y = W[(size_t)(k0 + 1) * DDIM + nt * 16 + lr];
            acc = wmma_f32(a, b, acc);
        }
        int col = nt * 16 + lr;
        float bgv = (wave == 3) ? bg[col] : 0.f;
        #pragma unroll
        for (int r = 0; r < 8; ++r) {
            int grow = row0 + hi * 8 + r;
            float val = acc[r];
            if (wave == 3) val = 1.f / (1.f + __expf(-(val + bgv)));
            out[(size_t)grow * DDIM + col] = val;
        }
    }
}

// ---------------------------------------------------------------------------
// Kernel 2: pair bias  bias[h][i][j] = sum_p pair[i][j][p] * Wp[p][h]
// grid = 256 blocks x 256 threads, one (i,j) per thread.
// ---------------------------------------------------------------------------
__global__ __launch_bounds__(256)
void pair_bias_kernel(const float* __restrict__ pair, const float* __restrict__ Wp,
                      float* __restrict__ bias)
{
    __shared__ float wps[256];
    if (threadIdx.x < 256) wps[threadIdx.x] = Wp[threadIdx.x];
    __syncthreads();

    int idx = blockIdx.x * 256 + threadIdx.x;   // i*256 + j
    const float4* p4 = (const float4*)(pair + (size_t)idx * 64);
    float acc0 = 0.f, acc1 = 0.f, acc2 = 0.f, acc3 = 0.f;
    #pragma unroll
    for (int c4 = 0; c4 < 16; ++c4) {
        float4 x = p4[c4];
        const float* w = &wps[c4 * 16];
        acc0 += x.x * w[0] + x.y * w[4] + x.z * w[8]  + x.w * w[12];
        acc1 += x.x * w[1] + x.y * w[5] + x.z * w[9]  + x.w * w[13];
        acc2 += x.x * w[2] + x.y * w[6] + x.z * w[10] + x.w * w[14];
        acc3 += x.x * w[3] + x.y * w[7] + x.z * w[11] + x.w * w[15];
    }
    bias[0 * 65536 + idx] = acc0;
    bias[1 * 65536 + idx] = acc1;
    bias[2 * 65536 + idx] = acc2;
    bias[3 * 65536 + idx] = acc3;
}

// ---------------------------------------------------------------------------
// Kernel 3: attention per (s,h).  grid = 512 blocks x 256 threads (8 waves).
// Each wave handles i-tiles {w, w+8}.  Full-row softmax over 256 keys.
// Q/K/V staging uses CDNA5 async global->LDS DMA (ASYNCcnt) when available.
// ---------------------------------------------------------------------------
__global__ __launch_bounds__(256)
void attn_kernel(const float* __restrict__ qg, const float* __restrict__ kg,
                 const float* __restrict__ vg, const float* __restrict__ gg,
                 const float* __restrict__ bias, const int* __restrict__ mask,
                 float* __restrict__ attn_out)
{
    __shared__ __align__(16) float Qs[256][20];   // stride 20: float4-aligned rows,
    __shared__ __align__(16) float Ks[256][20];   // gcd(20,64)=4 -> 16 distinct banks
    __shared__ __align__(16) float Vs[256][20];
    __shared__ float Pt[8][16][20];               // per-wave transpose scratch

    const int s   = blockIdx.x >> 2;
    const int h   = blockIdx.x & 3;
    const int tid = threadIdx.x;

    {   // stage Q/K/V head slices: one row per thread, async DMA to LDS
        const size_t base = (size_t)s * NDIM * DDIM + h * HD;
        int j = tid;
        const float* qs = qg + base + (size_t)j * DDIM;
        const float* ks = kg + base + (size_t)j * DDIM;
        const float* vs = vg + base + (size_t)j * DDIM;
        #pragma unroll
        for (int c = 0; c < 4; ++c) {
            async_copy16(qs + c * 4, &Qs[j][c * 4]);
            async_copy16(ks + c * 4, &Ks[j][c * 4]);
            async_copy16(vs + c * 4, &Vs[j][c * 4]);
        }
        async_copy_wait();
    }
    __syncthreads();

    const int wave = tid >> 5, lane = tid & 31;
    const int lr = lane & 15, hi = lane >> 4;

    for (int rep = 0; rep < 2; ++rep) {
        const int i0 = (wave + rep * 8) * 16;

        // ---- scores: 16 j-tiles of QK^T * 0.25 + bias, with mask ----
        v8f sc[16];
        #pragma unroll
        for (int jt = 0; jt < 16; ++jt) {
            v8f acc = {};
            #pragma unroll
            for (int kt = 0; kt < 4; ++kt) {
                int k0 = kt * 4 + hi * 2;
                v2f a; a.x = Qs[i0 + lr][k0];      a.y = Qs[i0 + lr][k0 + 1];
                v2f b; b.x = Ks[jt * 16 + lr][k0]; b.y = Ks[jt * 16 + lr][k0 + 1];
                acc = wmma_f32(a, b, acc);
            }
            int j  = jt * 16 + lr;
            int mv = mask[s * NDIM + j];
            const float* bp = bias + (size_t)h * 65536 + (size_t)(i0 + hi * 8) * NDIM + j;
            #pragma unroll
            for (int r = 0; r < 8; ++r) {
                float val = acc[r] * 0.25f + bp[(size_t)r * NDIM];
                acc[r] = (mv == 0) ? -INFINITY : val;
            }
            sc[jt] = acc;
        }

        // ---- row max (reduce over 16 tiles, then across 16-lane half) ----
        v8f mx = sc[0];
        #pragma unroll
        for (int jt = 1; jt < 16; ++jt)
            #pragma unroll
            for (int r = 0; r < 8; ++r) mx[r] = fmaxf(mx[r], sc[jt][r]);
        #pragma unroll
        for (int off = 1; off < 16; off <<= 1)
            #pragma unroll
            for (int r = 0; r < 8; ++r)
                mx[r] = fmaxf(mx[r], __shfl_xor(mx[r], off, 32));

        // ---- exp + row sum ----
        v8f sm = {};
        #pragma unroll
        for (int jt = 0; jt < 16; ++jt)
            #pragma unroll
            for (int r = 0; r < 8; ++r) {
                float e = __expf(sc[jt][r] - mx[r]);
                sc[jt][r] = e;
                sm[r] += e;
            }
        #pragma unroll
        for (int off = 1; off < 16; off <<= 1)
            #pragma unroll
            for (int r = 0; r < 8; ++r)
                sm[r] += __shfl_xor(sm[r], off, 32);

        // ---- O = P @ V  (transpose P through per-wave LDS tile) ----
        v8f o = {};
        for (int jt = 0; jt < 16; ++jt) {
            #pragma unroll
            for (int r = 0; r < 8; ++r)
                Pt[wave][hi * 8 + r][lr] = sc[jt][r];
            __syncthreads();     // uniform across all waves (rep/jt counts equal)
            #pragma unroll
            for (int kt = 0; kt < 4; ++kt) {
                int k0 = kt * 4 + hi * 2;
                v2f a; a.x = Pt[wave][lr][k0];      a.y = Pt[wave][lr][k0 + 1];
                v2f b; b.x = Vs[jt * 16 + k0][lr];  b.y = Vs[jt * 16 + k0 + 1][lr];
                o = wmma_f32(a, b, o);
            }
            __syncthreads();
        }

        // ---- normalize, gate, store ----
        #pragma unroll
        for (int r = 0; r < 8; ++r) {
            int i = i0 + hi * 8 + r;
            size_t oidx = (size_t)s * NDIM * DDIM + (size_t)i * DDIM + h * HD + lr;
            float gate = gg[oidx];
            attn_out[oidx] = (o[r] / sm[r]) * gate;
        }
    }
}

// ---------------------------------------------------------------------------
// Kernel 4: output projection  Y = attn(32768x64) @ Wo + bo.
// grid = 2048 blocks x 128 threads; wave w computes column tile w.
// ---------------------------------------------------------------------------
__global__ __launch_bounds__(128)
void out_proj_kernel(const float* __restrict__ a, const float* __restrict__ Wo,
                     const float* __restrict__ bo, float* __restrict__ out)
{
    const int tid = threadIdx.x;
    const int wave = tid >> 5, lane = tid & 31;
    const int lr = lane & 15, hi = lane >> 4;
    const int row0 = blockIdx.x * 16;
    const int nt = wave;

    v8f acc = {};
    #pragma unroll
    for (int kt = 0; kt < 16; ++kt) {
        int k0 = kt * 4 + hi * 2;
        float2 av2 = *(const float2*)(a + (size_t)(row0 + lr) * DDIM + k0);
        v2f av; av.x = av2.x; av.y = av2.y;
        v2f bv; bv.x = Wo[(size_t)k0 * DDIM + nt * 16 + lr];
                bv.y = Wo[(size_t)(k0 + 1) * DDIM + nt * 16 + lr];
        acc = wmma_f32(av, bv, acc);
    }
    int col = nt * 16 + lr;
    float bb = bo[col];
    #pragma unroll
    for (int r = 0; r < 8; ++r)
        out[(size_t)(row0 + hi * 8 + r) * DDIM + col] = acc[r] + bb;
}

// ---------------------------------------------------------------------------
extern "C" void kernel_launch(void* const* d_in, const int* in_sizes, int n_in,
                              void* d_out, int out_size, void* d_ws, size_t ws_size,
                              hipStream_t stream)
{
    const float* msa  = (const float*)d_in[0];
    const float* pair = (const float*)d_in[1];
    const int*   mask = (const int*)  d_in[2];
    const float* lng  = (const float*)d_in[3];
    const float* lnb  = (const float*)d_in[4];
    const float* Wq   = (const float*)d_in[5];
    const float* Wk   = (const float*)d_in[6];
    const float* Wv   = (const float*)d_in[7];
    const float* Wg   = (const float*)d_in[8];
    const float* bg   = (const float*)d_in[9];
    const float* Wp   = (const float*)d_in[10];
    const float* Wo   = (const float*)d_in[11];
    const float* bo   = (const float*)d_in[12];

    float* ws   = (float*)d_ws;
    float* q    = ws;
    float* k    = q    + (size_t)ROWS * DDIM;
    float* v    = k    + (size_t)ROWS * DDIM;
    float* g    = v    + (size_t)ROWS * DDIM;
    float* bias = g    + (size_t)ROWS * DDIM;
    float* attn = bias + (size_t)HEADS * NDIM * NDIM;

    ln_qkvg_kernel<<<dim3(ROWS / 16), dim3(128), 0, stream>>>(
        msa, lng, lnb, Wq, Wk, Wv, Wg, bg, q, k, v, g);
    pair_bias_kernel<<<dim3(NDIM * NDIM / 256), dim3(256), 0, stream>>>(
        pair, Wp, bias);
    attn_kernel<<<dim3(SDIM * HEADS), dim3(256), 0, stream>>>(
        q, k, v, g, bias, mask, attn);
    out_proj_kernel<<<dim3(ROWS / 16), dim3(128), 0, stream>>>(
        attn, Wo, bo, (float*)d_out);
}